// NKSSummary_17875653886471
// MI455X (gfx1250) — compile-verified
//
#include <hip/hip_runtime.h>
#include <hip/hip_bf16.h>

typedef float v2f __attribute__((ext_vector_type(2)));
typedef float v8f __attribute__((ext_vector_type(8)));

#define D_ 128
#define K_ 128

// ---------------------------------------------------------------------------
// Kernel A: baseline vectors. bev[d] = exp(lbev[d]);
// bar[d] = sum_{j>=d} (exp(lbev[j]) + exp(lbcc[j]))
// ---------------------------------------------------------------------------
__global__ __launch_bounds__(D_) void nks_baseline(
    const float* __restrict__ lbev, const float* __restrict__ lbcc,
    float* __restrict__ bev, float* __restrict__ bar) {
  __shared__ float s[D_];
  int d = threadIdx.x;
  float e = __expf(lbev[d]);
  float c = __expf(lbcc[d]);
  bev[d] = e;
  s[d] = e + c;
  __syncthreads();
  float acc = 0.0f;
  for (int j = d; j < D_; ++j) acc += s[j];
  bar[d] = acc;
}

// ---------------------------------------------------------------------------
// Kernel B: exemplar tables. EVt = exp(log_ev);  S = exp(log_ev)+exp(log_cc);
// ARt = S @ T with T[j,d] = (j >= d)  -> suffix cumsum, done as an f32 WMMA
// GEMM (V_WMMA_F32_16X16X4_F32). Triangular B generated in registers; chunks
// that are entirely below the tile's column origin are skipped (B == 0).
// One block = 16 exemplar rows; 4 waves each own two 16-col output tiles.
// ---------------------------------------------------------------------------
__global__ __launch_bounds__(128) void nks_tables(
    const float* __restrict__ log_ev, const float* __restrict__ log_cc,
    float* __restrict__ EVt, float* __restrict__ ARt, int E) {
  __shared__ float S[16][130];  // stride 130 -> bank-conflict-free A reads
  const int e0 = blockIdx.x * 16;
  const int tid = threadIdx.x;
  const bool full = (e0 + 16 <= E);  // uniform: tile fully in range

  // Stage S into LDS, write EV table to global.
  if (full) {
    for (int idx = tid; idx < 16 * D_; idx += 128) {
      int r = idx >> 7, col = idx & (D_ - 1);
      size_t g = (size_t)(e0 + r) * D_ + col;
      float ev = __expf(log_ev[g]);
      float cc = __expf(log_cc[g]);
      EVt[g] = ev;
      S[r][col] = ev + cc;
    }
  } else {
    for (int idx = tid; idx < 16 * D_; idx += 128) {
      int r = idx >> 7, col = idx & (D_ - 1);
      int e = e0 + r;
      int esafe = (e < E) ? e : (E - 1);
      size_t g = (size_t)esafe * D_ + col;
      float ev = __expf(log_ev[g]);
      float cc = __expf(log_cc[g]);
      if (e < E) EVt[g] = ev;
      S[r][col] = ev + cc;
    }
  }
  __syncthreads();

  const int wave = tid >> 5;
  const int lane = tid & 31;
  const int m = lane & 15;     // A row / B-C column within tile
  const int kh = lane >> 4;    // which K-pair this lane holds

  for (int t = 0; t < 2; ++t) {
    const int dt = wave + t * 4;  // d-tile index 0..7 (uniform per wave)
    const int n0 = dt * 16;       // tile column origin
    const int n = n0 + m;         // global output column for B/C
    v8f c = {0.f, 0.f, 0.f, 0.f, 0.f, 0.f, 0.f, 0.f};
    // Chunks kc < dt*4 have j0..j0+3 < n0 <= n for all lanes -> B == 0: skip.
#pragma unroll 4
    for (int kc = dt * 4; kc < 32; ++kc) {
      int j0 = kc * 4 + kh * 2;  // K indices held by this lane: j0, j0+1
      v2f a;
      a.x = S[m][j0];
      a.y = S[m][j0 + 1];
      v2f b;
      b.x = (j0 >= n) ? 1.0f : 0.0f;
      b.y = (j0 + 1 >= n) ? 1.0f : 0.0f;
      c = __builtin_amdgcn_wmma_f32_16x16x4_f32(
          /*neg_a=*/false, a, /*neg_b=*/false, b,
          /*c_mod=*/(short)0, c, /*reuse_a=*/false, /*reuse_b=*/false);
    }
    // C/D layout: VGPR r -> M = r (lanes 0-15) / r+8 (lanes 16-31), N = m.
    if (full) {
#pragma unroll
      for (int r = 0; r < 8; ++r) {
        ARt[(size_t)(e0 + r + kh * 8) * D_ + n] = c[r];
      }
    } else {
#pragma unroll
      for (int r = 0; r < 8; ++r) {
        int e = e0 + r + kh * 8;
        if (e < E) ARt[(size_t)e * D_ + n] = c[r];
      }
    }
  }
}

// ---------------------------------------------------------------------------
// Kernel C: gather phase. One wave per b-row; lane owns 4 of 128 d-columns.
// w[k], labels staged per-wave in LDS (no cross-wave sync needed: DS ops
// from the same wave are in-order). Tables (102 MB) live in the 192 MB L2.
// ---------------------------------------------------------------------------
__global__ __launch_bounds__(256) void nks_main(
    const float* __restrict__ sq_dists, const int* __restrict__ labels,
    const float* __restrict__ EVt, const float* __restrict__ ARt,
    const float* __restrict__ bev, const float* __restrict__ bar,
    float* __restrict__ out, int B) {
  __shared__ float w_s[8][K_];
  __shared__ int l_s[8][K_];
  const int tid = threadIdx.x;
  const int wave = tid >> 5;
  const int lane = tid & 31;
  const int b = blockIdx.x * 8 + wave;
  if (b >= B) return;  // wave-uniform exit

  const float* sq = sq_dists + (size_t)b * K_;
  const int* lb = labels + (size_t)b * K_;
#pragma unroll
  for (int i = 0; i < 4; ++i) {
    int k = lane * 4 + i;
    float s = sq[k];
    w_s[wave][k] = (s <= 2.0f) ? __expf(-s) : 0.0f;
    l_s[wave][k] = lb[k];
  }

  const int d0 = lane * 4;
  float nx = 0.f, ny = 0.f, nz = 0.f, nw = 0.f;
  float dx = 0.f, dy = 0.f, dz = 0.f, dw = 0.f;
#pragma unroll 4
  for (int k = 0; k < K_; ++k) {
    float wk = w_s[wave][k];
    size_t row = (size_t)l_s[wave][k] * D_;
    float4 ev = *(const float4*)(EVt + row + d0);
    float4 ar = *(const float4*)(ARt + row + d0);
    nx = fmaf(wk, ev.x, nx); ny = fmaf(wk, ev.y, ny);
    nz = fmaf(wk, ev.z, nz); nw = fmaf(wk, ev.w, nw);
    dx = fmaf(wk, ar.x, dx); dy = fmaf(wk, ar.y, dy);
    dz = fmaf(wk, ar.z, dz); dw = fmaf(wk, ar.w, dw);
  }

  float4 be = *(const float4*)(bev + d0);
  float4 ba = *(const float4*)(bar + d0);
  const float lo = 1e-12f, hi = 1.0f - 1e-12f, eps = 1e-12f;
  float4 o;
  o.x = fminf(fmaxf((nx + be.x) / (dx + ba.x + eps), lo), hi);
  o.y = fminf(fmaxf((ny + be.y) / (dy + ba.y + eps), lo), hi);
  o.z = fminf(fmaxf((nz + be.z) / (dz + ba.z + eps), lo), hi);
  o.w = fminf(fmaxf((nw + be.w) / (dw + ba.w + eps), lo), hi);
  *(float4*)(out + (size_t)b * D_ + d0) = o;
}

// ---------------------------------------------------------------------------
extern "C" void kernel_launch(void* const* d_in, const int* in_sizes, int n_in,
                              void* d_out, int out_size, void* d_ws,
                              size_t ws_size, hipStream_t stream) {
  const float* sq_dists = (const float*)d_in[0];
  const float* log_ev = (const float*)d_in[1];
  const float* log_cc = (const float*)d_in[2];
  const float* lbev = (const float*)d_in[3];
  const float* lbcc = (const float*)d_in[4];
  const int* labels = (const int*)d_in[5];
  float* out = (float*)d_out;

  const int D = in_sizes[3];       // 128
  const int E = in_sizes[1] / D;   // 100000
  const int B = out_size / D;      // 4096

  float* EVt = (float*)d_ws;                 // E*D floats (51.2 MB)
  float* ARt = EVt + (size_t)E * D;          // E*D floats (51.2 MB)
  float* bev = ARt + (size_t)E * D;          // D floats
  float* bar = bev + D;                      // D floats

  nks_baseline<<<1, D_, 0, stream>>>(lbev, lbcc, bev, bar);
  nks_tables<<<(E + 15) / 16, 128, 0, stream>>>(log_ev, log_cc, EVt, ARt, E);
  nks_main<<<(B + 7) / 8, 256, 0, stream>>>(sq_dists, labels, EVt, ARt, bev,
                                            bar, out, B);
}